// BalancedLoss_13769665151353
// MI455X (gfx1250) — compile-verified
//
#include <hip/hip_runtime.h>
#include <hip/hip_bf16.h>

// ---------------------------------------------------------------------------
// BalancedLoss (weighted BCE-with-logits mean) for pred/target [B=65536,C=512]
// Single streaming pass (HBM-bound, ~268MB @ 23.3TB/s => ~11.5us floor):
//   per column c accumulate: cnt = sum(target), s1 = sum(bce*target),
//   st = sum(bce).  Finalize derives majority/minority weights per column and
//   reduces 512 column contributions to one scalar.
// Stage A: deterministic per-block partials in workspace (no float atomics).
// Stage B: per-column combine + WMMA (V_WMMA_F32_16X16X4_F32, A = ones) for
//          the 512->16 column-sum reduction, shfl_xor for the last 16->1.
// ---------------------------------------------------------------------------

typedef float v2f __attribute__((ext_vector_type(2)));
typedef float v8f __attribute__((ext_vector_type(8)));

#define CCOLS 512
#define TPB_A 128           // each thread owns 4 contiguous columns (float4)
#define PART_STRIDE (3 * CCOLS)

__device__ __forceinline__ float bce_logits(float p, float t) {
    // logaddexp(0, p) - p*t  ==  max(p,0) + log1p(exp(-|p|)) - p*t
    return fmaxf(p, 0.0f) - p * t + log1pf(expf(-fabsf(p)));
}

__global__ void balanced_loss_partials(const float4* __restrict__ pred4,
                                       const float4* __restrict__ targ4,
                                       float* __restrict__ partials,
                                       int Brows, int rowsPerBlock) {
    const int t = threadIdx.x;                    // 0..127 -> cols 4t..4t+3
    const int r0 = blockIdx.x * rowsPerBlock;
    int r1 = r0 + rowsPerBlock;
    if (r1 > Brows) r1 = Brows;

    float4 cnt = make_float4(0.f, 0.f, 0.f, 0.f);
    float4 s1  = make_float4(0.f, 0.f, 0.f, 0.f);
    float4 st  = make_float4(0.f, 0.f, 0.f, 0.f);

    const int rowQ = CCOLS / 4;                   // float4s per row (=128)

    #pragma unroll 2
    for (int r = r0; r < r1; ++r) {
        const size_t idx = (size_t)r * rowQ + t;
        if (r + 16 < r1) {                        // speculative stream prefetch
            __builtin_prefetch(&pred4[idx + (size_t)16 * rowQ], 0, 1);
            __builtin_prefetch(&targ4[idx + (size_t)16 * rowQ], 0, 1);
        }
        const float4 p  = pred4[idx];             // global_load_b128
        const float4 tg = targ4[idx];             // global_load_b128

        float b;
        b = bce_logits(p.x, tg.x); cnt.x += tg.x; s1.x = fmaf(b, tg.x, s1.x); st.x += b;
        b = bce_logits(p.y, tg.y); cnt.y += tg.y; s1.y = fmaf(b, tg.y, s1.y); st.y += b;
        b = bce_logits(p.z, tg.z); cnt.z += tg.z; s1.z = fmaf(b, tg.z, s1.z); st.z += b;
        b = bce_logits(p.w, tg.w); cnt.w += tg.w; s1.w = fmaf(b, tg.w, s1.w); st.w += b;
    }

    // partials layout: [block][ {cnt[512], s1[512], st[512]} ]  (coalesced)
    float* base = partials + (size_t)blockIdx.x * PART_STRIDE;
    const int c0 = 4 * t;
    base[c0 + 0]             = cnt.x; base[c0 + 1]             = cnt.y;
    base[c0 + 2]             = cnt.z; base[c0 + 3]             = cnt.w;
    base[CCOLS + c0 + 0]     = s1.x;  base[CCOLS + c0 + 1]     = s1.y;
    base[CCOLS + c0 + 2]     = s1.z;  base[CCOLS + c0 + 3]     = s1.w;
    base[2 * CCOLS + c0 + 0] = st.x;  base[2 * CCOLS + c0 + 1] = st.y;
    base[2 * CCOLS + c0 + 2] = st.z;  base[2 * CCOLS + c0 + 3] = st.w;
}

__global__ void balanced_loss_finalize(const float* __restrict__ partials,
                                       const float* __restrict__ pos_prop,
                                       float* __restrict__ out,
                                       int nblocks, int Brows) {
    __shared__ float sCon[CCOLS];
    const int c = threadIdx.x;                    // 0..511

    float cnt = 0.f, s1 = 0.f, st = 0.f;
    for (int j = 0; j < nblocks; ++j) {           // coalesced across threads
        const float* base = partials + (size_t)j * PART_STRIDE;
        cnt += base[c];
        s1  += base[CCOLS + c];
        st  += base[2 * CCOLS + c];
    }

    const float Bf  = (float)Brows;
    const float bal = pos_prop[c] * Bf;           // balance_num
    const bool majIsOne = (cnt >= bal);
    const float n_maj = majIsOne ? cnt : (Bf - cnt);
    const float n_min = Bf - n_maj;
    const float S_maj = majIsOne ? s1 : (st - s1);
    const float S_min = st - S_maj;
    const float w_maj = bal / n_maj;              // n_maj >= bal > 0
    const float w_min = (n_min > 0.f) ? (Bf - bal) / fmaxf(n_min, 1.f) : 1.f;

    sCon[c] = w_maj * S_maj + w_min * S_min;      // per-column contribution
    __syncthreads();

    // ---- wave 0: view sCon as 32x16 row-major; sum all entries ----
    // 8 accumulating V_WMMA_F32_16X16X4_F32 with A = ones(16x4):
    //   D[m][n] += sum_k B[k][n]  -> every D row holds running column sums.
    // f32 B operand layout (wave32): lane L, VGPR v  ->  B[2*(L/16)+v][L%16]
    if (threadIdx.x < 32) {
        const int lane = threadIdx.x;
        const int half = lane >> 4;               // 0 or 1
        const int n    = lane & 15;

        v2f a; a.x = 1.0f; a.y = 1.0f;            // ones A-matrix
        v8f acc = {};
        #pragma unroll
        for (int tt = 0; tt < 8; ++tt) {          // rows 4*tt .. 4*tt+3
            const int k0 = 4 * tt + 2 * half;
            v2f b;
            b.x = sCon[(k0 + 0) * 16 + n];
            b.y = sCon[(k0 + 1) * 16 + n];
            acc = __builtin_amdgcn_wmma_f32_16x16x4_f32(
                false, a, false, b, (short)0, acc, false, false);
        }
        // lane L now holds colsum[L%16] (D rows identical); 16 -> 1 via shfl
        float v = acc[0];
        v += __shfl_xor(v, 1, 32);
        v += __shfl_xor(v, 2, 32);
        v += __shfl_xor(v, 4, 32);
        v += __shfl_xor(v, 8, 32);
        if (lane == 0) out[0] = v / (Bf * (float)CCOLS);
    }
}

extern "C" void kernel_launch(void* const* d_in, const int* in_sizes, int n_in,
                              void* d_out, int out_size, void* d_ws, size_t ws_size,
                              hipStream_t stream) {
    const float* pred     = (const float*)d_in[0];
    const float* targ     = (const float*)d_in[1];
    const float* pos_prop = (const float*)d_in[2];
    float* out = (float*)d_out;
    float* partials = (float*)d_ws;

    const int total = in_sizes[0];
    const int Brows = total / CCOLS;              // C fixed at 512

    // Size the grid to the workspace: each block needs 3*512 floats of ws.
    const size_t perBlockBytes = (size_t)PART_STRIDE * sizeof(float);
    int nblocks = 1024;                           // ~64 rows/block at B=65536
    const int maxBlocks = (int)(ws_size / perBlockBytes);
    if (nblocks > maxBlocks) nblocks = maxBlocks;
    if (nblocks > Brows)     nblocks = Brows;
    if (nblocks < 1)         nblocks = 1;
    const int rpb = (Brows + nblocks - 1) / nblocks;
    nblocks = (Brows + rpb - 1) / rpb;            // exact cover, no empty blocks

    balanced_loss_partials<<<nblocks, TPB_A, 0, stream>>>(
        (const float4*)pred, (const float4*)targ, partials, Brows, rpb);

    balanced_loss_finalize<<<1, CCOLS, 0, stream>>>(
        partials, pos_prop, out, nblocks, Brows);
}